// DetectionLoss_29283087024791
// MI455X (gfx1250) — compile-verified
//
#include <hip/hip_runtime.h>

// Problem constants (match reference)
#define Bn 32
#define Cn 7
#define Hn 400
#define Wn 400
#define Tn 64
#define HWn (Hn * Wn)

typedef float v2f __attribute__((ext_vector_type(2)));
typedef float v8f __attribute__((ext_vector_type(8)));

// One block per batch image; one thread per target.
// Sparse evaluation: only the <=64 scattered cells per image contribute.
__global__ __launch_bounds__(64) void det_loss_partial(
    const float* __restrict__ preds,     // [B,C,H,W]
    const float* __restrict__ targets,   // [B,T,C]
    float* __restrict__ part)            // part[0..31] = sums, part[32..63] = counts
{
    const int b    = blockIdx.x;
    const int t    = threadIdx.x;   // 0..63
    const int lane = t & 31;

    __shared__ int   cellS[Tn];
    __shared__ float psumS[Tn];
    __shared__ int   scnt[2];

    const float* tgt = targets + ((size_t)b * Tn + t) * Cn;
    const float tx = tgt[0];
    const float ty = tgt[1];
    // floor then clip, matching reference (scale = W/80 = H/80 = 5.0 exactly)
    const int gx = (int)fminf(fmaxf(floorf(tx * 5.0f), 0.0f), (float)(Wn - 1));
    const int gy = (int)fminf(fmaxf(floorf(ty * 5.0f), 0.0f), (float)(Hn - 1));
    const int cell = gy * Wn + gx;
    cellS[t] = cell;

    // Issue prefetches for the 7 scattered channel values (7 distinct cachelines,
    // stride 640KB) so they are in flight while we do the dedup scan.
    const float* p = preds + (size_t)b * (Cn * HWn) + cell;
#pragma unroll
    for (int c = 0; c < Cn; ++c)
        __builtin_prefetch(p + (size_t)c * HWn, 0, 3);   // -> global_prefetch_b8

    __syncthreads();

    // Last-writer-wins dedup: target t survives iff no later target maps to same cell.
    bool winner = true;
    for (int u = t + 1; u < Tn; ++u)
        winner &= (cellS[u] != cell);

    float local = 0.0f;
    if (winner) {
#pragma unroll
        for (int c = 0; c < Cn; ++c) {
            float d  = p[(size_t)c * HWn] - tgt[c];
            float ad = fabsf(d);
            local += (ad < 1.0f) ? 0.5f * d * d : (ad - 0.5f);
        }
    }
    psumS[t] = local;

    // distinct-cell count (num_objects contribution), wave32 ballot per wave
    unsigned long long bal = __ballot(winner);
    if (lane == 0) scnt[t >> 5] = __popcll(bal);

    __syncthreads();

    // 64-way block reduction via V_WMMA_F32_16X16X4_F32 with B = ones:
    // A (16x4 f32, 2 VGPRs/lane) holds the 64 partials; D[m,n] = row-sum(A[m,:]).
    // Summing the 8 D VGPRs per lane yields rows 0-7 (lanes 0-15) / rows 8-15
    // (lanes 16-31); one shfl_xor(16) completes the total. f32 end to end.
    v2f a;
    a.x = psumS[2 * lane];
    a.y = psumS[2 * lane + 1];
    v2f ones;
    ones.x = 1.0f;
    ones.y = 1.0f;
    v8f cacc = {};
    v8f dmat = __builtin_amdgcn_wmma_f32_16x16x4_f32(
        /*neg_a=*/false, a, /*neg_b=*/false, ones,
        /*c_mod=*/(short)0, cacc, /*reuse_a=*/false, /*reuse_b=*/false);

    float s = dmat[0] + dmat[1] + dmat[2] + dmat[3] +
              dmat[4] + dmat[5] + dmat[6] + dmat[7];
    s += __shfl_xor(s, 16, 32);

    if (t == 0) {
        part[b]      = s;
        part[Bn + b] = (float)(scnt[0] + scnt[1]);
    }
}

// Single-wave finalize: fold 32 per-batch partials, emit (loss, num_objects).
__global__ __launch_bounds__(32) void det_loss_finalize(
    const float* __restrict__ part,
    float* __restrict__ out)
{
    const int lane = threadIdx.x;  // 0..31
    float s = part[lane];
    float c = part[Bn + lane];
#pragma unroll
    for (int off = 16; off > 0; off >>= 1) {
        s += __shfl_down(s, off, 32);
        c += __shfl_down(c, off, 32);
    }
    if (lane == 0) {
        out[0] = s / (c + 1e-6f);  // loss
        out[1] = c;                // num_objects (float32, == mask.sum())
    }
}

extern "C" void kernel_launch(void* const* d_in, const int* in_sizes, int n_in,
                              void* d_out, int out_size, void* d_ws, size_t ws_size,
                              hipStream_t stream) {
    const float* preds   = (const float*)d_in[0];   // [32,7,400,400] f32
    const float* targets = (const float*)d_in[1];   // [32,64,7] f32
    float* out  = (float*)d_out;                    // [loss, num_objects]
    float* part = (float*)d_ws;                     // 64 floats scratch (fully written each call)

    det_loss_partial<<<Bn, Tn, 0, stream>>>(preds, targets, part);
    det_loss_finalize<<<1, 32, 0, stream>>>(part, out);
}